// AttentionBlock_11029476016815
// MI455X (gfx1250) — compile-verified
//
#include <hip/hip_runtime.h>
#include <math.h>

#define NB   64
#define CCH  128
#define HWN  1024
#define CHW  (CCH * HWN)          // 131072
#define EPSV 1e-5f
#define ISQ  0.08838834764831845f // 1/sqrt(128)

typedef __attribute__((ext_vector_type(2))) float v2f;
typedef __attribute__((ext_vector_type(4))) float v4f;
typedef __attribute__((ext_vector_type(8))) float v8f;

__device__ __forceinline__ v8f zero8() {
  v8f z = {0.f, 0.f, 0.f, 0.f, 0.f, 0.f, 0.f, 0.f};
  return z;
}

// D(16x16,f32) = A(16x4,f32) x B(4x16,f32) + C, fp32 WMMA
__device__ __forceinline__ v8f wmma4(v2f a, v2f b, v8f c) {
  return __builtin_amdgcn_wmma_f32_16x16x4_f32(false, a, false, b, (short)0, c,
                                               false, false);
}

// ---------------------------------------------------------------------------
// Kernel 1: per-(tensor, sample) mean / rstd over C*H*W elements
// ---------------------------------------------------------------------------
__global__ __launch_bounds__(256) void ln_stats_kernel(
    const float* __restrict__ q, const float* __restrict__ k,
    const float* __restrict__ v, float* __restrict__ stats) {
  const int n = blockIdx.x, t = blockIdx.y, tid = threadIdx.x;
  const float* x = (t == 0 ? q : (t == 1 ? k : v)) + (size_t)n * CHW;
  const float4* x4 = (const float4*)x;
  float s = 0.f, ss = 0.f;
  for (int i = tid; i < CHW / 4; i += 256) {
    float4 u = x4[i];
    s += u.x + u.y + u.z + u.w;
    ss += u.x * u.x + u.y * u.y + u.z * u.z + u.w * u.w;
  }
  __shared__ float r1[256], r2[256];
  r1[tid] = s; r2[tid] = ss;
  __syncthreads();
  for (int off = 128; off > 0; off >>= 1) {
    if (tid < off) { r1[tid] += r1[tid + off]; r2[tid] += r2[tid + off]; }
    __syncthreads();
  }
  if (tid == 0) {
    float mean = r1[0] * (1.0f / (float)CHW);
    float var  = r2[0] * (1.0f / (float)CHW) - mean * mean;
    stats[(t * NB + n) * 2 + 0] = mean;
    stats[(t * NB + n) * 2 + 1] = 1.0f / sqrtf(var + EPSV);
  }
}

// ---------------------------------------------------------------------------
// Kernel 2: fused LayerNorm + 1x1 conv projection via fp32 WMMA.
// Block tile: M=128 (all out channels) x N=128 spatial, K=128.
// Grid: (1024/128, N, 3 tensors). 8 waves, each 32x64 (2x4 16x16 tiles).
// ---------------------------------------------------------------------------
__global__ __launch_bounds__(256) void proj_kernel(
    const float* __restrict__ query, const float* __restrict__ key_,
    const float* __restrict__ value,
    const float* __restrict__ ln1w, const float* __restrict__ ln1b,
    const float* __restrict__ ln2w, const float* __restrict__ ln2b,
    const float* __restrict__ ln3w, const float* __restrict__ ln3b,
    const float* __restrict__ wq, const float* __restrict__ bq,
    const float* __restrict__ wk, const float* __restrict__ bk,
    const float* __restrict__ wv, const float* __restrict__ bv,
    const float* __restrict__ stats,
    float* __restrict__ qo, float* __restrict__ ko, float* __restrict__ vo) {
  const int hw0 = blockIdx.x * 128;
  const int n   = blockIdx.y;
  const int t   = blockIdx.z;
  const int tid = threadIdx.x;

  const float* x    = t == 0 ? query : (t == 1 ? key_ : value);
  const float* lw   = t == 0 ? ln1w  : (t == 1 ? ln2w : ln3w);
  const float* lb   = t == 0 ? ln1b  : (t == 1 ? ln2b : ln3b);
  const float* W    = t == 0 ? wq    : (t == 1 ? wk   : wv);
  const float* bias = t == 0 ? bq    : (t == 1 ? bk   : bv);
  float* outp       = t == 0 ? qo    : (t == 1 ? ko   : vo);
  const float mu = stats[(t * NB + n) * 2 + 0];
  const float rs = stats[(t * NB + n) * 2 + 1];

  __shared__ __align__(16) float Wt[128 * 132]; // W transposed: [cin][cout], padded
  __shared__ __align__(16) float Xs[128 * 132]; // normalized X: [cin][j], padded

  for (int idx = tid; idx < 128 * 128; idx += 256) {
    int m = idx >> 7, c = idx & 127;
    Wt[c * 132 + m] = W[idx];
  }
  const size_t xbase = (size_t)n * CHW;
  for (int idx = tid; idx < 128 * 128; idx += 256) {
    int c = idx >> 7, j = idx & 127;
    int g = c * HWN + hw0 + j;
    Xs[c * 132 + j] = (x[xbase + g] - mu) * rs * lw[g] + lb[g];
  }
  __syncthreads();

  const int wave = tid >> 5, lane = tid & 31;
  const int lane16 = lane & 15, lh = lane >> 4;
  const int wm = (wave >> 1) * 32; // 0,32,64,96
  const int wn = (wave & 1) * 64;  // 0,64

  v8f acc[2][4];
#pragma unroll
  for (int i = 0; i < 2; ++i)
#pragma unroll
    for (int j = 0; j < 4; ++j) acc[i][j] = zero8();

  for (int c0 = 0; c0 < 128; c0 += 4) {
    const int ca = c0 + 2 * lh; // lanes 0-15: K=c0,c0+1 ; lanes 16-31: K=c0+2,c0+3
    v2f a[2], b[4];
#pragma unroll
    for (int mt = 0; mt < 2; ++mt) {
      a[mt].x = Wt[ca * 132 + wm + 16 * mt + lane16];
      a[mt].y = Wt[(ca + 1) * 132 + wm + 16 * mt + lane16];
    }
#pragma unroll
    for (int nt = 0; nt < 4; ++nt) {
      b[nt].x = Xs[ca * 132 + wn + 16 * nt + lane16];
      b[nt].y = Xs[(ca + 1) * 132 + wn + 16 * nt + lane16];
    }
#pragma unroll
    for (int mt = 0; mt < 2; ++mt)
#pragma unroll
      for (int nt = 0; nt < 4; ++nt)
        acc[mt][nt] = wmma4(a[mt], b[nt], acc[mt][nt]);
  }

#pragma unroll
  for (int mt = 0; mt < 2; ++mt)
#pragma unroll
    for (int nt = 0; nt < 4; ++nt)
#pragma unroll
      for (int r = 0; r < 8; ++r) {
        int m = wm + 16 * mt + 8 * lh + r;             // D layout: M=r / r+8
        int col = hw0 + wn + 16 * nt + lane16;         // N = lane%16
        outp[xbase + (size_t)m * HWN + col] = acc[mt][nt][r] + bias[m];
      }
}

// ---------------------------------------------------------------------------
// Kernel 3: fused attention per (n, 16-column j-tile).
//   S = K^T Q (1024 x 16) via WMMA  -> LDS (transposed, padded)
//   softmax over i (exact, max-shifted) in LDS
//   O = V * P via WMMA, V staged transposed through LDS in 64-row blocks
//   epilogue: out = qn (recomputed) + O
// ---------------------------------------------------------------------------
__global__ __launch_bounds__(256) void attn_kernel(
    const float* __restrict__ qp, const float* __restrict__ kp,
    const float* __restrict__ vp, const float* __restrict__ query,
    const float* __restrict__ ln1w, const float* __restrict__ ln1b,
    const float* __restrict__ stats, float* __restrict__ out) {
  const int j0 = blockIdx.x * 16;
  const int n  = blockIdx.y;
  const int tid = threadIdx.x;
  const int wave = tid >> 5, lane = tid & 31;
  const int lane16 = lane & 15, lh = lane >> 4;
  const size_t base = (size_t)n * CHW;

  __shared__ __align__(16) float Qs[128 * 16];   // Q tile [c][j]
  __shared__ __align__(16) float St[16 * 1028];  // S^T [j][i], padded stride
  __shared__ __align__(16) float Vt[64 * 132];   // V block transposed [i][c]
  __shared__ __align__(16) float red[16 * 16];

  for (int idx = tid; idx < 128 * 16; idx += 256) {
    int c = idx >> 4, jj = idx & 15;
    Qs[idx] = qp[base + c * HWN + j0 + jj];
  }
  __syncthreads();

  // ---- GEMM1: wave w computes S rows i in [128w, 128w+128), all 16 j ----
  {
    const int wi = wave * 128;
    v8f s[8];
#pragma unroll
    for (int it = 0; it < 8; ++it) s[it] = zero8();

    for (int c0 = 0; c0 < 128; c0 += 4) {
      const int ca = c0 + 2 * lh;
      v2f b;
      b.x = Qs[ca * 16 + lane16];
      b.y = Qs[(ca + 1) * 16 + lane16];
#pragma unroll
      for (int it = 0; it < 8; ++it) {
        size_t ga = base + (size_t)ca * HWN + wi + 16 * it + lane16;
        v2f a;
        a.x = kp[ga];        // K[ca][i]
        a.y = kp[ga + HWN];  // K[ca+1][i]
        s[it] = wmma4(a, b, s[it]);
      }
    }
#pragma unroll
    for (int it = 0; it < 8; ++it) {
      // lane holds S[i = wi+16*it + 8*lh + r][j = lane16], r=0..7 contiguous in i
      float* p = &St[lane16 * 1028 + wi + 16 * it + 8 * lh];
      v4f lo = {s[it][0], s[it][1], s[it][2], s[it][3]};
      v4f hi = {s[it][4], s[it][5], s[it][6], s[it][7]};
      *(v4f*)(p)     = lo;
      *(v4f*)(p + 4) = hi;
    }
  }
  __syncthreads();

  // ---- softmax over i for each column j; 16 threads per column ----
  {
    const int jl = tid & 15, grp = tid >> 4;
    float* row = &St[jl * 1028 + grp * 64];
    float m = -3.4e38f;
    for (int ii = 0; ii < 64; ++ii) m = fmaxf(m, row[ii]);
    red[grp * 16 + jl] = m;
    __syncthreads();
    float mg = -3.4e38f;
#pragma unroll
    for (int g = 0; g < 16; ++g) mg = fmaxf(mg, red[g * 16 + jl]);
    __syncthreads();
    float lsum = 0.f;
    for (int ii = 0; ii < 64; ++ii) {
      float e = expf(ISQ * (row[ii] - mg)); // scale by 1/sqrt(C) inside exp
      row[ii] = e;
      lsum += e;
    }
    red[grp * 16 + jl] = lsum;
    __syncthreads();
    float l = 0.f;
#pragma unroll
    for (int g = 0; g < 16; ++g) l += red[g * 16 + jl];
    float linv = 1.0f / l;
    for (int ii = 0; ii < 64; ++ii) row[ii] *= linv;
  }
  __syncthreads();

  // ---- GEMM2: wave w computes O rows c in [16w, 16w+16), k over i=1024 ----
  v8f o = zero8();
  const int c0w = wave * 16;
  for (int ib = 0; ib < 16; ++ib) {
    for (int idx = tid; idx < 64 * 128; idx += 256) {
      int c = idx >> 6, ii = idx & 63;
      Vt[ii * 132 + c] = vp[base + (size_t)c * HWN + ib * 64 + ii];
    }
    __syncthreads();
    for (int k0 = 0; k0 < 64; k0 += 4) {
      const int ka = k0 + 2 * lh;
      v2f a;
      a.x = Vt[ka * 132 + c0w + lane16];
      a.y = Vt[(ka + 1) * 132 + c0w + lane16];
      v2f b = *(const v2f*)&St[lane16 * 1028 + ib * 64 + ka]; // P[k][j], P[k+1][j]... via layout symmetry
      // note: b.x = P[ka][j], b.y = P[ka+1][j] (contiguous in i in St)
      o = wmma4(a, b, o);
    }
    __syncthreads();
  }

  // ---- epilogue: residual with recomputed normalized query ----
  const float muq = stats[n * 2 + 0];
  const float rsq = stats[n * 2 + 1];
#pragma unroll
  for (int r = 0; r < 8; ++r) {
    int c = c0w + 8 * lh + r;
    int g = c * HWN + j0 + lane16;
    float qn = (query[base + g] - muq) * rsq * ln1w[g] + ln1b[g];
    out[base + g] = qn + o[r];
  }
}

// ---------------------------------------------------------------------------
extern "C" void kernel_launch(void* const* d_in, const int* in_sizes, int n_in,
                              void* d_out, int out_size, void* d_ws,
                              size_t ws_size, hipStream_t stream) {
  const float* query = (const float*)d_in[0];
  const float* key_  = (const float*)d_in[1];
  const float* value = (const float*)d_in[2];
  const float* ln1w  = (const float*)d_in[3];
  const float* ln1b  = (const float*)d_in[4];
  const float* ln2w  = (const float*)d_in[5];
  const float* ln2b  = (const float*)d_in[6];
  const float* ln3w  = (const float*)d_in[7];
  const float* ln3b  = (const float*)d_in[8];
  const float* wq    = (const float*)d_in[9];
  const float* bq    = (const float*)d_in[10];
  const float* wk    = (const float*)d_in[11];
  const float* bk    = (const float*)d_in[12];
  const float* wv    = (const float*)d_in[13];
  const float* bv    = (const float*)d_in[14];

  float* ws    = (float*)d_ws;
  float* qbuf  = ws;
  float* kbuf  = ws + (size_t)NB * CHW;
  float* vbuf  = ws + 2 * (size_t)NB * CHW;
  float* stats = ws + 3 * (size_t)NB * CHW;
  float* out   = (float*)d_out;

  ln_stats_kernel<<<dim3(NB, 3), 256, 0, stream>>>(query, key_, value, stats);
  proj_kernel<<<dim3(HWN / 128, NB, 3), 256, 0, stream>>>(
      query, key_, value, ln1w, ln1b, ln2w, ln2b, ln3w, ln3b, wq, bq, wk, bk,
      wv, bv, stats, qbuf, kbuf, vbuf);
  attn_kernel<<<dim3(HWN / 16, NB), 256, 0, stream>>>(qbuf, kbuf, vbuf, query,
                                                      ln1w, ln1b, stats, out);
}